// POS_82643760709649
// MI455X (gfx1250) — compile-verified
//
#include <hip/hip_runtime.h>

// Fused compact+MLP, fp32 WMMA (V_WMMA_F32_16X16X4_F32).
// 2 M-tiles per wave: B-fragments (W1/W2 rows) shared across both, halving
// L2 traffic for weights; 14 independent WMMA chains per k-step.
// B=32, L=512, D=768, H=100 (pad 112), C=45 (pad 48).

typedef __attribute__((ext_vector_type(2))) float v2f;
typedef __attribute__((ext_vector_type(8))) float v8f;

#define BB   32
#define LL   512
#define DD   768
#define HH   100
#define CC   45
#define HPAD 112
#define NT1  (HPAD / 16)   // 7 N-tiles for GEMM1
#define NT2  3             // 3 N-tiles for GEMM2 (48 >= 45)
#define WAVES 4
#define MROWS 32           // rows per wave (2 M-tiles)

// ---------------------------------------------------------------------------
// Kernel 1: per-batch stable-compaction gather table + zero row.
// ---------------------------------------------------------------------------
__global__ __launch_bounds__(LL) void build_gather_kernel(
    const int* __restrict__ valid, int* __restrict__ gidx,
    float* __restrict__ zrow) {
  __shared__ int s[LL];
  const int b = blockIdx.x;
  const int t = threadIdx.x;
  if (b == 0) {
    for (int i = t; i < DD; i += LL) zrow[i] = 0.0f;
  }
  const int v = (valid[b * LL + t] == 1) ? 1 : 0;
  s[t] = v;
  __syncthreads();
  for (int off = 1; off < LL; off <<= 1) {  // Hillis-Steele inclusive scan
    int x = (t >= off) ? s[t - off] : 0;
    __syncthreads();
    s[t] += x;
    __syncthreads();
  }
  const int pos = s[t] - 1;
  gidx[b * LL + t] = -1;
  __syncthreads();
  if (v) gidx[b * LL + pos] = t;
}

// ---------------------------------------------------------------------------
// Kernel 2: fused  out = (gather(seq) @ W1^T + b1) @ W2^T + b2
// One wave per 32-row (2x16) M-block. k-loop outermost, B frags shared by
// both M-tiles, double-buffered loads, branch-free guards (pointer select
// for gathered A rows, index clamp for padded N columns).
// ---------------------------------------------------------------------------
__global__ __launch_bounds__(WAVES * 32) void fused_mlp_kernel(
    const float* __restrict__ seq, const int* __restrict__ gidx,
    const float* __restrict__ zrow,
    const float* __restrict__ W1, const float* __restrict__ b1,
    const float* __restrict__ W2, const float* __restrict__ b2,
    float* __restrict__ out) {
  __shared__ __align__(16) float hbuf[WAVES][MROWS * HPAD];

  const int lane = threadIdx.x & 31;
  const int wv   = threadIdx.x >> 5;
  const int blk  = blockIdx.x * WAVES + wv;  // 0 .. B*L/32-1
  const int b    = blk >> 4;                 // 16 M-blocks per batch
  const int m0   = (blk & 15) * MROWS;       // row offset inside batch
  const int hi   = lane >> 4;
  const int ln   = lane & 15;
  const int koff = 2 * hi;

  // Gathered A rows for the two M-tiles (zero row for invalid slots).
  const int r0 = gidx[b * LL + m0 + ln];
  const int r1 = gidx[b * LL + m0 + 16 + ln];
  const float* arow0 = (r0 >= 0) ? (seq + ((size_t)b * LL + r0) * DD) : zrow;
  const float* arow1 = (r1 >= 0) ? (seq + ((size_t)b * LL + r1) * DD) : zrow;

  // ---------------- GEMM1: h[32 x HPAD] = A @ W1^T + b1 --------------------
  int ncl1[NT1];
#pragma unroll
  for (int nt = 0; nt < NT1; ++nt) {
    int n = nt * 16 + ln;
    ncl1[nt] = (n < HH) ? n : (HH - 1);  // clamp: pad cols never read
  }

  v8f acc0[NT1], acc1[NT1];
#pragma unroll
  for (int nt = 0; nt < NT1; ++nt) { acc0[nt] = (v8f){}; acc1[nt] = (v8f){}; }

  // Double-buffered pipeline.
  v2f a0_c = *(const v2f*)(arow0 + koff);
  v2f a1_c = *(const v2f*)(arow1 + koff);
  v2f w_c[NT1];
#pragma unroll
  for (int nt = 0; nt < NT1; ++nt)
    w_c[nt] = *(const v2f*)(W1 + ncl1[nt] * DD + koff);

#pragma unroll 2
  for (int kk = 0; kk < DD; kk += 4) {
    const int kn = ((kk + 4 < DD) ? (kk + 4) : 0) + koff;  // wrap: harmless
    v2f a0_n = *(const v2f*)(arow0 + kn);
    v2f a1_n = *(const v2f*)(arow1 + kn);
    v2f w_n[NT1];
#pragma unroll
    for (int nt = 0; nt < NT1; ++nt)
      w_n[nt] = *(const v2f*)(W1 + ncl1[nt] * DD + kn);
#pragma unroll
    for (int nt = 0; nt < NT1; ++nt) {
      acc0[nt] = __builtin_amdgcn_wmma_f32_16x16x4_f32(
          false, a0_c, false, w_c[nt], (short)0, acc0[nt], false, false);
      acc1[nt] = __builtin_amdgcn_wmma_f32_16x16x4_f32(
          false, a1_c, false, w_c[nt], (short)0, acc1[nt], false, false);
    }
    a0_c = a0_n;
    a1_c = a1_n;
#pragma unroll
    for (int nt = 0; nt < NT1; ++nt) w_c[nt] = w_n[nt];
  }

  // Bias + stage both h tiles to LDS (row-major, stride HPAD).
  {
    float* hb = &hbuf[wv][0];
#pragma unroll
    for (int nt = 0; nt < NT1; ++nt) {
      const int n = nt * 16 + ln;
      const float bias = b1[ncl1[nt]];
#pragma unroll
      for (int i = 0; i < 8; ++i) {
        const int m = i + 8 * hi;
        hb[m * HPAD + n] = acc0[nt][i] + bias;
        hb[(m + 16) * HPAD + n] = acc1[nt][i] + bias;
      }
    }
  }

  // ---------------- GEMM2: out[32 x 45] = h[:, :100] @ W2^T + b2 -----------
  int ncl2[NT2];
#pragma unroll
  for (int nt = 0; nt < NT2; ++nt) {
    int n = nt * 16 + ln;
    ncl2[nt] = (n < CC) ? n : (CC - 1);
  }

  v8f c0[NT2], c1[NT2];
#pragma unroll
  for (int nt = 0; nt < NT2; ++nt) { c0[nt] = (v8f){}; c1[nt] = (v8f){}; }

  const float* hr0 = &hbuf[wv][ln * HPAD];         // A-frag row, M-tile 0
  const float* hr1 = &hbuf[wv][(ln + 16) * HPAD];  // A-frag row, M-tile 1
#pragma unroll 5
  for (int kk = 0; kk < HH; kk += 4) {
    const int k = kk + koff;                       // max 99 < HH
    v2f a0 = *(const v2f*)(hr0 + k);
    v2f a1 = *(const v2f*)(hr1 + k);
    v2f w[NT2];
#pragma unroll
    for (int nt = 0; nt < NT2; ++nt)
      w[nt] = *(const v2f*)(W2 + ncl2[nt] * HH + k);
#pragma unroll
    for (int nt = 0; nt < NT2; ++nt) {
      c0[nt] = __builtin_amdgcn_wmma_f32_16x16x4_f32(
          false, a0, false, w[nt], (short)0, c0[nt], false, false);
      c1[nt] = __builtin_amdgcn_wmma_f32_16x16x4_f32(
          false, a1, false, w[nt], (short)0, c1[nt], false, false);
    }
  }

#pragma unroll
  for (int nt = 0; nt < NT2; ++nt) {
    const int n = nt * 16 + ln;
    if (n < CC) {                                  // divergence only on stores
      const float bias = b2[n];
#pragma unroll
      for (int i = 0; i < 8; ++i) {
        const int m = i + 8 * hi;
        out[((size_t)b * LL + m0 + m) * CC + n] = c0[nt][i] + bias;
        out[((size_t)b * LL + m0 + 16 + m) * CC + n] = c1[nt][i] + bias;
      }
    }
  }
}

// ---------------------------------------------------------------------------
extern "C" void kernel_launch(void* const* d_in, const int* in_sizes, int n_in,
                              void* d_out, int out_size, void* d_ws,
                              size_t ws_size, hipStream_t stream) {
  (void)in_sizes; (void)n_in; (void)out_size; (void)ws_size;
  const float* seq   = (const float*)d_in[0];
  const int*   valid = (const int*)d_in[1];
  const float* W1    = (const float*)d_in[2];
  const float* b1    = (const float*)d_in[3];
  const float* W2    = (const float*)d_in[4];
  const float* b2    = (const float*)d_in[5];
  float*       out   = (float*)d_out;

  int*   gidx = (int*)d_ws;                                     // 64 KB
  float* zrow = (float*)((char*)d_ws + BB * LL * sizeof(int));  // 768 floats

  build_gather_kernel<<<BB, LL, 0, stream>>>(valid, gidx, zrow);

  const int mblocks = (BB * LL) / MROWS;  // 512 waves
  fused_mlp_kernel<<<mblocks / WAVES, WAVES * 32, 0, stream>>>(
      seq, gidx, zrow, W1, b1, W2, b2, out);
}